// RepNorm1d_31791347925866
// MI455X (gfx1250) — compile-verified
//
#include <hip/hip_runtime.h>

#define RDIM 2048          // rep dimension (in_sizes[1])
#define CDIM 512           // channels (in_sizes[2])
#define NB_P1 1024         // blocks in pass 1
#define RPB 4              // rows per pipeline buffer in pass 1
#define NB_SA 64           // stage-A stats blocks (NB_P1 / 16)
#define EPS_NORM 1e-4f
#define EPS_REP  0.01f
#define BN_EPS   1e-5f

#if defined(__HIP_DEVICE_COMPILE__) && defined(__gfx1250__)
#define USE_ASYNC 1
#else
#define USE_ASYNC 0
#endif

typedef __attribute__((address_space(3))) float lds_f;
typedef float v4f __attribute__((ext_vector_type(4)));   // native vector for NT builtins

// Issue one 8KB row load: 512 threads x b128 = 2048 floats, memory -> LDS,
// tracked by ASYNCcnt. NT hint: `ten` is streamed exactly once, keep it out
// of L2 so the x0 working set (102 MB < 192 MB L2) stays resident for pass 3.
__device__ __forceinline__ void row_issue(const float* __restrict__ g,
                                          float* lds, int tid) {
#if USE_ASYNC
  unsigned lds_off =
      (unsigned)(unsigned long long)(lds_f*)(lds + tid * 4);
  unsigned long long ga = (unsigned long long)(g + tid * 4);
  asm volatile("global_load_async_to_lds_b128 %0, %1, off th:TH_LOAD_NT"
               :: "v"(lds_off), "v"(ga) : "memory");
#else
  ((v4f*)(lds))[tid] = ((const v4f*)(g))[tid];
#endif
}

// Wait until at most one chunk of async loads is outstanding (this wave).
__device__ __forceinline__ void row_wait(bool chunk_pending) {
#if USE_ASYNC
  if (chunk_pending) {
    asm volatile("s_wait_asynccnt 0x4" ::: "memory");  // next chunk (4) in flight
  } else {
    asm volatile("s_wait_asynccnt 0x0" ::: "memory");
  }
#endif
}

// Build segment starts from rep_idx (non-decreasing). segstart has CDIM+1 ints.
__global__ void rn_seg(const int* __restrict__ rep_idx,
                       int* __restrict__ segstart, int rdim, int cdim) {
  int r = blockIdx.x * blockDim.x + threadIdx.x;
  if (r < rdim) {
    if (r == 0 || rep_idx[r] != rep_idx[r - 1]) segstart[rep_idx[r]] = r;
    if (r == 0) segstart[cdim] = rdim;
  }
}

// Pass 1: per-row norm contraction + x2 output + block partial sums for BN.
// 4-row chunks, double-buffered via async-to-LDS; 3 barriers per 32 KB.
__global__ __launch_bounds__(512) void rn_pass1(
    const float* __restrict__ ten, const int* __restrict__ rep_idx,
    const float* __restrict__ beta, const int* __restrict__ segstart,
    float* __restrict__ x0_out, float* __restrict__ x2_out,
    float* __restrict__ partial, int n_rows, int rows_per_block) {
  __shared__ __align__(16) float buf[2][RPB][RDIM];  // 2 x 32 KB
  __shared__ float recip[RPB][CDIM];
  __shared__ int sidx[RDIM];

  const int tid = threadIdx.x;
  const int b = blockIdx.x;
  const int row0 = b * rows_per_block;
  const int row1 = min(n_rows, row0 + rows_per_block);

  for (int r = tid; r < RDIM; r += 512) sidx[r] = rep_idx[r];

  const int c = tid;                   // one channel per thread
  const int s = segstart[c];
  const int e = segstart[c + 1];
  const float bet = beta[c];
  float acc = 0.0f, acc2 = 0.0f;

  const int nchunks = (row1 > row0) ? (row1 - row0 + RPB - 1) / RPB : 0;

  // Prologue: issue chunk 0 (tail rows clamp to last valid row -> fixed
  // issue count of RPB keeps the asynccnt wait threshold a literal).
  if (nchunks > 0) {
    const int base = row0;
    for (int r = 0; r < RPB; ++r) {
      const int row = min(base + r, row1 - 1);
      row_issue(ten + (size_t)row * RDIM, buf[0][r], tid);
    }
  }
  __syncthreads();                     // sidx visible

  for (int ch = 0; ch < nchunks; ++ch) {
    const int cur = ch & 1;
    const bool more = (ch + 1 < nchunks);
    if (more) {
      const int nbase = row0 + (ch + 1) * RPB;
      for (int r = 0; r < RPB; ++r) {
        const int row = min(nbase + r, row1 - 1);
        row_issue(ten + (size_t)row * RDIM, buf[cur ^ 1][r], tid);
      }
    }
    row_wait(more);                    // chunk ch complete (in-order retire)
    __syncthreads();

    const int base = row0 + ch * RPB;
    const int cnt = min(RPB, row1 - base);

    // Phase A: per-channel segment reduction, x0, recip for each row.
    for (int r = 0; r < cnt; ++r) {
      float ss = 0.0f;
      for (int k = s; k < e; ++k) {
        const float v = buf[cur][r][k];
        ss = fmaf(v, v, ss);
      }
      const float x0 = fmaxf(sqrtf(ss), EPS_NORM);
      x0_out[(size_t)(base + r) * CDIM + c] = x0;   // RT store: keep in L2
      acc += x0;
      acc2 = fmaf(x0, x0, acc2);
      recip[r][c] = 1.0f / (fabsf(x0 + bet) + EPS_REP);
    }
    __syncthreads();

    // Phase B: x2 rows straight from LDS, streaming NT stores.
    for (int r = 0; r < cnt; ++r) {
      const v4f tv = ((const v4f*)buf[cur][r])[tid];
      const int r4 = tid * 4;
      v4f o;
      o.x = tv.x * recip[r][sidx[r4 + 0]];
      o.y = tv.y * recip[r][sidx[r4 + 1]];
      o.z = tv.z * recip[r][sidx[r4 + 2]];
      o.w = tv.w * recip[r][sidx[r4 + 3]];
      __builtin_nontemporal_store(
          o, ((v4f*)(x2_out + (size_t)(base + r) * RDIM)) + tid);
    }
    __syncthreads();                   // protect buf[cur] before reuse
  }

  partial[(size_t)b * (2 * CDIM) + c] = acc;
  partial[(size_t)b * (2 * CDIM) + CDIM + c] = acc2;
}

// Stats stage A: 64 blocks, each folds 16 block-partials (coalesced per lane).
__global__ __launch_bounds__(CDIM) void rn_stats_a(
    const float* __restrict__ partial, float* __restrict__ partialB) {
  const int c = threadIdx.x;
  const int g = blockIdx.x;
  float s = 0.0f, s2 = 0.0f;
  for (int p = g * 16; p < g * 16 + 16; ++p) {
    s += partial[(size_t)p * (2 * CDIM) + c];
    s2 += partial[(size_t)p * (2 * CDIM) + CDIM + c];
  }
  partialB[(size_t)g * (2 * CDIM) + c] = s;
  partialB[(size_t)g * (2 * CDIM) + CDIM + c] = s2;
}

// Stats stage B: final fold (double) -> mean & rsqrt(var+eps).
__global__ __launch_bounds__(CDIM) void rn_stats_b(
    const float* __restrict__ partialB, float* __restrict__ stats,
    int nblocks, float inv_n) {
  const int c = threadIdx.x;
  double s = 0.0, s2 = 0.0;
  for (int p = 0; p < nblocks; ++p) {
    s += (double)partialB[(size_t)p * (2 * CDIM) + c];
    s2 += (double)partialB[(size_t)p * (2 * CDIM) + CDIM + c];
  }
  const double mean = s * (double)inv_n;
  const double var = s2 * (double)inv_n - mean * mean;
  stats[c] = (float)mean;
  stats[CDIM + c] = (float)rsqrt(var + (double)BN_EPS);
}

// Pass 3: in-place batchnorm apply on the x1 region (holds x0 from pass 1,
// expected to hit L2). Data is dead after this: last-use loads, NT stores.
__global__ __launch_bounds__(256) void rn_bn(
    float* __restrict__ x01, const float* __restrict__ stats, size_t n4) {
  __shared__ float smean[CDIM];
  __shared__ float sinv[CDIM];
  for (int i = threadIdx.x; i < CDIM; i += blockDim.x) {
    smean[i] = stats[i];
    sinv[i] = stats[CDIM + i];
  }
  __syncthreads();
  const size_t stride = (size_t)gridDim.x * blockDim.x;
  for (size_t i = (size_t)blockIdx.x * blockDim.x + threadIdx.x; i < n4;
       i += stride) {
    v4f v = __builtin_nontemporal_load(((const v4f*)x01) + i);
    const int cb = (int)((i % (CDIM / 4)) * 4);
    v.x = (v.x - smean[cb + 0]) * sinv[cb + 0];
    v.y = (v.y - smean[cb + 1]) * sinv[cb + 1];
    v.z = (v.z - smean[cb + 2]) * sinv[cb + 2];
    v.w = (v.w - smean[cb + 3]) * sinv[cb + 3];
    __builtin_nontemporal_store(v, ((v4f*)x01) + i);
  }
}

extern "C" void kernel_launch(void* const* d_in, const int* in_sizes, int n_in,
                              void* d_out, int out_size, void* d_ws,
                              size_t ws_size, hipStream_t stream) {
  const float* ten = (const float*)d_in[0];      // [N, 2048]
  const int* rep_idx = (const int*)d_in[1];      // [2048]
  const float* beta = (const float*)d_in[2];     // [512]

  const int rdim = in_sizes[1];                  // 2048
  const int n = in_sizes[0] / rdim;              // 50000

  float* out = (float*)d_out;
  float* x1 = out;                               // [N, 512]
  float* x2 = out + (size_t)n * CDIM;            // [N, 2048]

  // Workspace: segstart (4KB pad) | partial (4MB) | partialB (256KB) | stats
  int* segstart = (int*)d_ws;
  float* partial = (float*)((char*)d_ws + 4096);
  float* partialB = partial + (size_t)NB_P1 * 2 * CDIM;
  float* stats = partialB + (size_t)NB_SA * 2 * CDIM;

  rn_seg<<<(rdim + 255) / 256, 256, 0, stream>>>(rep_idx, segstart, rdim, CDIM);

  const int rpb = (n + NB_P1 - 1) / NB_P1;
  rn_pass1<<<NB_P1, 512, 0, stream>>>(ten, rep_idx, beta, segstart, x1, x2,
                                      partial, n, rpb);

  rn_stats_a<<<NB_SA, CDIM, 0, stream>>>(partial, partialB);
  rn_stats_b<<<1, CDIM, 0, stream>>>(partialB, stats, NB_SA, 1.0f / (float)n);

  const size_t n4 = (size_t)n * CDIM / 4;
  rn_bn<<<2048, 256, 0, stream>>>(x1, stats, n4);
}